// FeatureContraction_47004122087963
// MI455X (gfx1250) — compile-verified
//
#include <hip/hip_runtime.h>

typedef __attribute__((ext_vector_type(2))) float v2f;
typedef __attribute__((ext_vector_type(8))) float v8f;

// Problem dims (from reference): B=128, C=128, X=3, Y=16
//   x    : [B, C, X, Y, Y, Y]  fp32   (rows = X*Y*Y = 768 per (b,c), K = Y = 16)
//   attr : [B, C, Y]           fp32
//   out  : [B, C, X, Y, Y]     fp32
#define BC_TOTAL (128 * 128)
#define KDIM 16
#define ROWS 768          // X*Y*Y rows per (b,c)
#define TILES (ROWS / 16) // 48 row-tiles of 16 per (b,c)
#define WAVES_PER_BLOCK 8

__global__ __launch_bounds__(256) void FeatureContraction_47004122087963_kernel(
    const float* __restrict__ x,
    const float* __restrict__ attr,
    float* __restrict__ out)
{
    const int lane = threadIdx.x & 31;
    const int wave = threadIdx.x >> 5;
    const int bc   = blockIdx.x * WAVES_PER_BLOCK + wave; // uniform per wave
    if (bc >= BC_TOTAL) return;

    const int rowN = lane & 15; // B-matrix column = output row within tile
    const int half = lane >> 4; // K half: lanes 0-15 -> K={0,1}, 16-31 -> K={2,3}

    // ---- A fragments: attr vector broadcast down the M dimension ----
    // A is 16x4 f32: lane m%16, VGPR0 = K=(half?2:0), VGPR1 = K+1.
    // A[m][k] = a[4q + k]  (independent of m)  -> one aligned b64 load per chunk.
    const float* abase = attr + (size_t)bc * KDIM;
    v2f afrag[4];
#pragma unroll
    for (int q = 0; q < 4; ++q)
        afrag[q] = *(const v2f*)(abase + q * 4 + half * 2);

    const float* xbase = x + (size_t)bc * (size_t)ROWS * KDIM;
    float*       obase = out + (size_t)bc * ROWS;

    for (int t = 0; t < TILES; ++t) {
        // B fragments: B[k][n] = x[row = t*16 + n, i = 4q + k]
        // lane layout: N = lane%16, K pair selected by lane half -> b64 loads.
        const float* xl = xbase + (size_t)(t * 16 + rowN) * KDIM + half * 2;

        // Prefetch next tile's slice for this lane (global_prefetch_b8).
        if (t + 1 < TILES)
            __builtin_prefetch(xl + 16 * KDIM, 0, 1);

        v2f bfrag[4];
#pragma unroll
        for (int q = 0; q < 4; ++q)
            bfrag[q] = *(const v2f*)(xl + q * 4);

        // D = A x B + C, chained over the K=16 contraction (4 chunks of 4).
        // Every row m of D is identical: D[m][n] = sum_i a[i] * x[row=n][i].
        v8f acc = {};
#pragma unroll
        for (int q = 0; q < 4; ++q) {
            acc = __builtin_amdgcn_wmma_f32_16x16x4_f32(
                /*neg_a=*/false, afrag[q],
                /*neg_b=*/false, bfrag[q],
                /*c_mod=*/(short)0, acc,
                /*reuse_a=*/false, /*reuse_b=*/false);
        }

        // Lane L (<16): VGPR0 of D = D[M=0][N=L] = out[row = t*16 + L].
        // Lanes 16-31 hold duplicates (M=8..15 rows) -> mask them off.
        if (half == 0)
            obase[t * 16 + rowN] = acc[0];
    }
}

extern "C" void kernel_launch(void* const* d_in, const int* in_sizes, int n_in,
                              void* d_out, int out_size, void* d_ws, size_t ws_size,
                              hipStream_t stream)
{
    (void)in_sizes; (void)n_in; (void)out_size; (void)d_ws; (void)ws_size;
    const float* x    = (const float*)d_in[0];
    const float* attr = (const float*)d_in[1];
    float*       out  = (float*)d_out;

    const int blocks = BC_TOTAL / WAVES_PER_BLOCK; // 2048 blocks x 8 waves = 16384 waves
    FeatureContraction_47004122087963_kernel<<<blocks, 256, 0, stream>>>(x, attr, out);
}